// Kinematics_for_yumi_20057497272555
// MI455X (gfx1250) — compile-verified
//
#include <hip/hip_runtime.h>
#include <stdint.h>
#include <math.h>

typedef float    v2f  __attribute__((ext_vector_type(2)));
typedef float    v8f  __attribute__((ext_vector_type(8)));
typedef uint32_t v4u  __attribute__((ext_vector_type(4)));
typedef int      v8i  __attribute__((ext_vector_type(8)));
typedef int      v4i  __attribute__((ext_vector_type(4)));

#define NJ      7
#define EPB     256     // elements per block
#define WPB     8       // waves per block
#define GROUPS  8       // 4-element chain groups per wave (8 * 4 * 8 waves = 256)

// Compile-time-only reordering fence: same-wave LDS is in-order (DScnt "in order"),
// so cross-lane store->load inside one wave needs no hardware wait, only a compiler fence.
#define COMPILER_FENCE() asm volatile("" ::: "memory")

// Per-(lane-class, step) operands for the B matrix build, precomputed on the HOST in
// double precision.  Lane class = khalf*4 + ir (8 classes).  For its two K rows the
// lane computes   b = (ir<2) ? c*x + s*y : z.
struct OpTab { float x0, y0, z0, x1, y1, z1; };
struct KTab  { OpTab t[8][NJ]; };

__global__ __launch_bounds__(256)
void yumi_fk_wmma(const float* __restrict__ eulers, float* __restrict__ out, int B,
                  KTab ktab)
{
    __shared__ float sAng[EPB * NJ];     // TDM-staged block tile of joint angles
    __shared__ float sD[WPB][16 * 20];   // per-wave D staging, padded stride 20 (bank-safe)

    const int tid  = threadIdx.x;
    const int lane = tid & 31;
    const int wave = tid >> 5;
    const long long blockBase = (long long)blockIdx.x * EPB;

    // --- TDM: async DMA of this block's 256x7 angle tile into LDS (wave 0 issues once) ---
    if (wave == 0) {
        uint64_t ga = (uint64_t)(uintptr_t)(eulers + blockBase * NJ);
        uint32_t la = (uint32_t)(uintptr_t)&sAng[0];   // low 32 bits of flat LDS addr = LDS offset
        const uint32_t nf = EPB * NJ;                   // 1792 f32 elements, 1-D tile
        v4u g0; v8i g1; v4i gz = {0, 0, 0, 0};
        g0[0] = 1u;                                               // count=1 (valid descriptor)
        g0[1] = la;                                               // lds_addr [63:32]
        g0[2] = (uint32_t)ga;                                     // global_addr[31:0]
        g0[3] = (uint32_t)((ga >> 32) & 0x01FFFFFFu) | (2u << 30);// addr[56:32] | type=2
        g1[0] = (int)(2u << 16);                                  // data_size=4B
        g1[1] = (int)((nf & 0xFFFFu) << 16);                      // tensor_dim0[15:0]
        g1[2] = (int)((nf >> 16) & 0xFFFFu) | (1 << 16);          // tensor_dim0[31:16] | tensor_dim1=1
        g1[3] = (int)((nf & 0xFFFFu) << 16);                      // tile_dim0 = 1792
        g1[4] = 0;                                                // tile_dim1/2 unused -> 1-D
        g1[5] = (int)nf;                                          // tensor_dim0_stride[31:0]
        g1[6] = 0;
        g1[7] = 0;
#if __clang_major__ >= 23
        v8i gz8 = {0, 0, 0, 0, 0, 0, 0, 0};
        __builtin_amdgcn_tensor_load_to_lds(g0, g1, gz, gz, gz8, 0);
#else
        __builtin_amdgcn_tensor_load_to_lds(g0, g1, gz, gz, 0);
#endif
        __builtin_amdgcn_s_wait_tensorcnt(0);
    }

    // Lane roles (fixed for the whole kernel):
    //   A operand (16x4 f32): lane m / m+16 hold row M=m, K={0,1} / K={2,3}  (ISA 7.12.2)
    //   B operand (4x16 f32): lane n / n+16 hold col N=n, K={0,1} / K={2,3}  (mirrored layout)
    float* sDw = &sD[wave][0];
    const int m     = lane & 15;    // stacked row (A) == stacked col (B)
    const int khalf = lane >> 4;    // 0 -> K={0,1}, 1 -> K={2,3}
    const int ib    = m >> 2;       // which of the 4 batch elems in the group
    const int ir    = m & 3;        // row within 4x4 (A) == col within 4x4 (B)
    const bool dynsel = (ir < 2);

    // per-lane operand table (vector loads from the kernarg segment; overlap with TDM)
    float xA[NJ], yA[NJ], zA[NJ], xB[NJ], yB[NJ], zB[NJ];
    {
        const OpTab* mt = ktab.t[khalf * 4 + ir];
#pragma unroll
        for (int stp = 0; stp < NJ; ++stp) {
            xA[stp] = mt[stp].x0; yA[stp] = mt[stp].y0; zA[stp] = mt[stp].z0;
            xB[stp] = mt[stp].x1; yB[stp] = mt[stp].y1; zB[stp] = mt[stp].z1;
        }
    }
    const float a0_init = (ir == 2 * khalf)     ? 1.f : 0.f;   // T_0 = I
    const float a1_init = (ir == 2 * khalf + 1) ? 1.f : 0.f;

    // output lane mapping (fixed): stacked row / column-pair this lane emits
    const int row16 = lane >> 1;
    const int cp    = (lane & 1) * 2;
    const int bo    = row16 >> 2;
    const int ro    = row16 & 3;
    const int ovIdx = row16 * 20 + 4 * bo + cp;          // LDS gather index for output
    const int naIdx = m * 20 + 4 * ib + 2 * khalf;       // LDS gather index for next A

    __syncthreads();   // sAng ready

    for (int g = 0; g < GROUPS; ++g) {
        const int e0 = wave * 32 + g * 4;   // local base element of this 4-element group

        // one sincos per lane covers all 28 (joint, elem) pairs of the group
        const int pl = (lane < 28) ? lane : 27;
        const float ang = sAng[(e0 + (pl & 3)) * NJ + (pl >> 2)];
        float s_l, c_l;
        __sincosf(ang, &s_l, &c_l);
        const int ci = __float_as_int(c_l);
        const int si = __float_as_int(s_l);

        // ---- pre-build ALL 7 B operands (independent of the chain) ----
        float b0v[NJ], b1v[NJ];
#pragma unroll
        for (int stp = 0; stp < NJ; ++stp) {
            const int src = (stp * 4 + ib) << 2;
            const float c = __int_as_float(__builtin_amdgcn_ds_bpermute(src, ci));
            const float s = __int_as_float(__builtin_amdgcn_ds_bpermute(src, si));
            const float dA = c * xA[stp] + s * yA[stp];
            const float dB = c * xB[stp] + s * yB[stp];
            b0v[stp] = dynsel ? dA : zA[stp];
            b1v[stp] = dynsel ? dB : zB[stp];
        }

        // one 64-bit address per group; per-step offset folds into the store immediate
        float* outg = out + (size_t)(blockBase + e0 + bo) * 112 + ro * 4 + cp;
        const bool gok = (blockBase + e0 + 4) <= (long long)B;   // wave-uniform guard

        // ---- serial chain: wmma -> LDS scatter -> LDS gather -> wmma ----
        float a0 = a0_init, a1 = a1_init;
#pragma unroll
        for (int stp = 0; stp < NJ; ++stp) {
            v2f A;  A.x  = a0;        A.y  = a1;
            v2f Bm; Bm.x = b0v[stp];  Bm.y = b1v[stp];
            v8f Cz = {0.f, 0.f, 0.f, 0.f, 0.f, 0.f, 0.f, 0.f};
            // D(16x16) = A(16x4) x B(4x16): 4 independent T_prev @ M products on the diagonal
            v8f D = __builtin_amdgcn_wmma_f32_16x16x4_f32(
                        false, A, false, Bm, (short)0, Cz, false, false);

            // stage D row-major into this wave's LDS tile (VGPR r holds M = r + 8*khalf, N = m)
#pragma unroll
            for (int r = 0; r < 8; ++r)
                sDw[(khalf * 8 + r) * 20 + m] = D[r];
            COMPILER_FENCE();   // keep the cross-lane gathers below after the scatters above

            // next-step A = diagonal 4x4 blocks of D; output = same blocks, row-pair view
            a0 = sDw[naIdx + 0];
            a1 = sDw[naIdx + 1];
            v2f ov;
            ov.x = sDw[ovIdx + 0];
            ov.y = sDw[ovIdx + 1];
            COMPILER_FENCE();   // keep next iteration's scatters after these gathers

            if (gok)
                __builtin_nontemporal_store(ov, (v2f*)(outg + stp * 16));
        }
    }
}

extern "C" void kernel_launch(void* const* d_in, const int* in_sizes, int n_in,
                              void* d_out, int out_size, void* d_ws, size_t ws_size,
                              hipStream_t stream) {
    (void)n_in; (void)out_size; (void)d_ws; (void)ws_size;

    // YUMI left-arm static joint origins (URDF): T_static = Txyz * Rz(y)*Ry(p)*Rx(r),
    // composed here on the host in double precision each call (deterministic).
    static const double tr[NJ][3] = {
        { 0.05355, 0.0725, 0.41492}, { 0.03, 0.0, 0.1 }, {-0.03, 0.17283, 0.0},
        {-0.04188, 0.0, 0.07873},    { 0.0405, 0.16461, 0.0},
        {-0.027, 0.0, 0.10039},      { 0.027, 0.029, 0.0},
    };
    static const double rpy[NJ][3] = {
        {-0.9795, -0.5682, 2.3155},     { 1.5707963, 0.0, 0.0}, {-1.5707963, 0.0, 0.0},
        { 1.5707963, -1.5707963, 0.0},  {-1.5707963, 0.0, 0.0},
        { 1.5707963, 0.0, 0.0},         {-1.5707963, 0.0, 0.0},
    };

    double T[NJ][4][4];
    for (int j = 0; j < NJ; ++j) {
        double r = rpy[j][0], p = rpy[j][1], y = rpy[j][2];
        double cr = cos(r), sr = sin(r), cp = cos(p), sp = sin(p), cy = cos(y), sy = sin(y);
        T[j][0][0] = cy * cp; T[j][0][1] = cy * sp * sr - sy * cr; T[j][0][2] = cy * sp * cr + sy * sr;
        T[j][1][0] = sy * cp; T[j][1][1] = sy * sp * sr + cy * cr; T[j][1][2] = sy * sp * cr - cy * sr;
        T[j][2][0] = -sp;     T[j][2][1] = cp * sr;                T[j][2][2] = cp * cr;
        T[j][0][3] = tr[j][0]; T[j][1][3] = tr[j][1]; T[j][2][3] = tr[j][2];
        T[j][3][0] = 0.0; T[j][3][1] = 0.0; T[j][3][2] = 0.0; T[j][3][3] = 1.0;
    }

    // Per-lane-class operand table: class = khalf*4 + ir, rows K = 2*khalf, 2*khalf+1.
    //   ir==0: x=u, y=v (col0 of Ts@Rz)   ir==1: x=v, y=-u (col1)   ir==2: z=w   ir==3: z=t
    KTab ktab;
    for (int khalf = 0; khalf < 2; ++khalf) {
        for (int ir = 0; ir < 4; ++ir) {
            for (int stp = 0; stp < NJ; ++stp) {
                const int k0 = 2 * khalf;
                const double u0 = T[stp][k0][0],     v0 = T[stp][k0][1];
                const double w0 = T[stp][k0][2],     t0 = T[stp][k0][3];
                const double u1 = T[stp][k0 + 1][0], v1 = T[stp][k0 + 1][1];
                const double w1 = T[stp][k0 + 1][2], t1 = T[stp][k0 + 1][3];
                OpTab& o = ktab.t[khalf * 4 + ir][stp];
                o.x0 = (float)((ir == 0) ? u0 : v0);
                o.y0 = (float)((ir == 0) ? v0 : -u0);
                o.z0 = (float)((ir == 2) ? w0 : t0);
                o.x1 = (float)((ir == 0) ? u1 : v1);
                o.y1 = (float)((ir == 0) ? v1 : -u1);
                o.z1 = (float)((ir == 2) ? w1 : t1);
            }
        }
    }

    const float* eulers = (const float*)d_in[0];
    float* out = (float*)d_out;
    const int B = in_sizes[0] / NJ;                 // 262144
    const int blocks = (B + EPB - 1) / EPB;         // 1024 (exact for the harness shape)
    yumi_fk_wmma<<<dim3(blocks), dim3(256), 0, stream>>>(eulers, out, B, ktab);
}